// GIN_17171279249890
// MI455X (gfx1250) — compile-verified
//
#include <hip/hip_runtime.h>
#include <hip/hip_bf16.h>

#define N_NODES 65536
#define N_EDGES 1048576
#define N_GRAPHS 512
#define IN_DIM 128
#define HID 64
#define N_LAYERS 4
#define N_CLASS 10
#define TSTRIDE 68  // padded row stride for the per-wave intermediate tile (bank-conflict-free)

typedef __attribute__((ext_vector_type(2))) float v2f;
typedef __attribute__((ext_vector_type(8))) float v8f;

// Swizzled LDS index for weight element W[r][c] so that a B-fragment
// {W[k+2h][n*16+col], W[k+2h+1][n*16+col]} is one contiguous float2:
//   dst = (r>>2)*256 + (c>>4)*64 + (c&15)*4 + ((r>>1)&1)*2 + (r&1)
// Fragment load address: (k>>2)*256 + n*64 + col*4 + half*2
// -> 32 lanes cover 64 consecutive dwords = 64 distinct LDS banks.
__device__ __forceinline__ int wswz(int r, int c) {
    return (r >> 2) * 256 + (c >> 4) * 64 + (c & 15) * 4 + ((r >> 1) & 1) * 2 + (r & 1);
}

// ---------------- utility kernels ----------------
__global__ void k_zero(float* __restrict__ p, int n) {
    int i = blockIdx.x * blockDim.x + threadIdx.x;
    int stride = gridDim.x * blockDim.x;
    for (; i < n; i += stride) p[i] = 0.0f;
}

__global__ void k_copy4(float4* __restrict__ dst, const float4* __restrict__ src, int n4) {
    int i = blockIdx.x * blockDim.x + threadIdx.x;
    int stride = gridDim.x * blockDim.x;
    for (; i < n4; i += stride) dst[i] = src[i];
}

// ---------------- fc1: h = x @ fc1_w + fc1_b  ([65536,128]x[128,64]) ----------------
// 8 waves/block, each wave computes a 16x64 strip with v_wmma_f32_16x16x4_f32.
__global__ __launch_bounds__(256) void k_fc1(const float* __restrict__ x,
                                             const float* __restrict__ w,     // [128][64]
                                             const float* __restrict__ bias,  // [64]
                                             float* __restrict__ h) {
    __shared__ float wl[IN_DIM * HID];  // swizzled, 32KB
    for (int i = threadIdx.x; i < IN_DIM * HID; i += 256)
        wl[wswz(i >> 6, i & 63)] = w[i];
    __syncthreads();

    const int wave    = threadIdx.x >> 5;
    const int lane    = threadIdx.x & 31;
    const int col     = lane & 15;
    const int half    = lane >> 4;
    const int rowBase = blockIdx.x * 128 + wave * 16;
    const int arow    = rowBase + col;
    const int bOff    = col * 4 + half * 2;

    v8f acc[4];
#pragma unroll
    for (int n = 0; n < 4; ++n) acc[n] = (v8f){0.f,0.f,0.f,0.f,0.f,0.f,0.f,0.f};

    for (int k = 0; k < IN_DIM; k += 4) {
        v2f a = *(const v2f*)(x + arow * IN_DIM + k + 2 * half);
        const float* wk = wl + (k >> 2) * 256 + bOff;
#pragma unroll
        for (int n = 0; n < 4; ++n) {
            v2f b = *(const v2f*)(wk + n * 64);
            acc[n] = __builtin_amdgcn_wmma_f32_16x16x4_f32(
                false, a, false, b, (short)0, acc[n], false, false);
        }
    }
#pragma unroll
    for (int n = 0; n < 4; ++n) {
        const int c  = n * 16 + col;
        const float bb = bias[c];
#pragma unroll
        for (int v = 0; v < 8; ++v) {
            const int row = rowBase + v + 8 * half;
            h[row * HID + c] = acc[n][v] + bb;
        }
    }
}

// ---------------- edge aggregation: agg[dst] += h[src] ----------------
// 16 threads/edge, float4 each, native f32 atomics resolving in L2 (h fits in 192MB L2).
__global__ __launch_bounds__(256) void k_edges(const int* __restrict__ esrc,
                                               const int* __restrict__ edst,
                                               const float* __restrict__ h,
                                               float* __restrict__ agg) {
    const int t = blockIdx.x * blockDim.x + threadIdx.x;
    const int e = t >> 4;
    if (e >= N_EDGES) return;
    const int d = (t & 15) * 4;
    const int s  = esrc[e];
    const int dd = edst[e];
    const float4 v = *(const float4*)(h + s * HID + d);
    float* p = agg + dd * HID + d;
    unsafeAtomicAdd(p + 0, v.x);
    unsafeAtomicAdd(p + 1, v.y);
    unsafeAtomicAdd(p + 2, v.z);
    unsafeAtomicAdd(p + 3, v.w);
}

// ---------------- fused GIN MLP layer ----------------
// hout = relu( (relu(agg@W1+b1)@W2 + b2) * n_norm ) + xin
__global__ __launch_bounds__(256) void k_mlp(const float* __restrict__ agg,
                                             const float* __restrict__ xin,
                                             const float* __restrict__ W1,  // [64][64]
                                             const float* __restrict__ b1,  // [64]
                                             const float* __restrict__ W2,  // [64][64]
                                             const float* __restrict__ b2,  // [64]
                                             const float* __restrict__ n_norm,
                                             float* __restrict__ hout) {
    __shared__ float w1l[HID * HID];        // swizzled
    __shared__ float w2l[HID * HID];        // swizzled
    __shared__ float tbuf[8][16 * TSTRIDE]; // per-wave 16x64 intermediate, padded stride

    for (int i = threadIdx.x; i < HID * HID; i += 256) {
        const int d = wswz(i >> 6, i & 63);
        w1l[d] = W1[i];
        w2l[d] = W2[i];
    }
    __syncthreads();

    const int wave    = threadIdx.x >> 5;
    const int lane    = threadIdx.x & 31;
    const int col     = lane & 15;
    const int half    = lane >> 4;
    const int rowBase = blockIdx.x * 128 + wave * 16;
    const int arow    = rowBase + col;
    const int bOff    = col * 4 + half * 2;
    float* t = tbuf[wave];

    // ---- phase 1: relu(agg @ W1 + b1) -> LDS tile ----
    v8f acc[4];
#pragma unroll
    for (int n = 0; n < 4; ++n) acc[n] = (v8f){0.f,0.f,0.f,0.f,0.f,0.f,0.f,0.f};

    for (int k = 0; k < HID; k += 4) {
        v2f a = *(const v2f*)(agg + arow * HID + k + 2 * half);
        const float* wk = w1l + (k >> 2) * 256 + bOff;
#pragma unroll
        for (int n = 0; n < 4; ++n) {
            v2f b = *(const v2f*)(wk + n * 64);
            acc[n] = __builtin_amdgcn_wmma_f32_16x16x4_f32(
                false, a, false, b, (short)0, acc[n], false, false);
        }
    }
#pragma unroll
    for (int n = 0; n < 4; ++n) {
        const int c  = n * 16 + col;
        const float bb = b1[c];
#pragma unroll
        for (int v = 0; v < 8; ++v) {
            const int r = v + 8 * half;
            const float m = acc[n][v] + bb;
            t[r * TSTRIDE + c] = m > 0.0f ? m : 0.0f;
        }
    }
    // same-wave LDS producer/consumer; DScnt waits inserted by compiler

    // ---- phase 2: (t @ W2 + b2) * n_norm, relu, + residual ----
#pragma unroll
    for (int n = 0; n < 4; ++n) acc[n] = (v8f){0.f,0.f,0.f,0.f,0.f,0.f,0.f,0.f};

    for (int k = 0; k < HID; k += 4) {
        // bank index = k + 4*col + 2*half (mod 64): all 32 lanes distinct
        v2f a = *(const v2f*)(t + col * TSTRIDE + k + 2 * half);
        const float* wk = w2l + (k >> 2) * 256 + bOff;
#pragma unroll
        for (int n = 0; n < 4; ++n) {
            v2f b = *(const v2f*)(wk + n * 64);
            acc[n] = __builtin_amdgcn_wmma_f32_16x16x4_f32(
                false, a, false, b, (short)0, acc[n], false, false);
        }
    }
#pragma unroll
    for (int n = 0; n < 4; ++n) {
        const int c  = n * 16 + col;
        const float bb = b2[c];
#pragma unroll
        for (int v = 0; v < 8; ++v) {
            const int row = rowBase + v + 8 * half;
            float m = (acc[n][v] + bb) * n_norm[row];
            m = m > 0.0f ? m : 0.0f;
            hout[row * HID + c] = m + xin[row * HID + c];
        }
    }
}

// ---------------- per-graph sum pooling ----------------
__global__ __launch_bounds__(256) void k_pool(const float* __restrict__ feat,
                                              const int* __restrict__ indicator,
                                              float* __restrict__ pooled) {
    const int t = blockIdx.x * blockDim.x + threadIdx.x;
    const int node = t >> 4;
    if (node >= N_NODES) return;
    const int d = (t & 15) * 4;
    const int g = indicator[node];
    const float4 v = *(const float4*)(feat + node * HID + d);
    float* p = pooled + g * HID + d;
    unsafeAtomicAdd(p + 0, v.x);
    unsafeAtomicAdd(p + 1, v.y);
    unsafeAtomicAdd(p + 2, v.z);
    unsafeAtomicAdd(p + 3, v.w);
}

// ---------------- readout ----------------
__global__ void k_score(const float* __restrict__ pooled,   // [5][512][64]
                        const float* __restrict__ sw,       // [5][64][10]
                        const float* __restrict__ sb,       // [5][10]
                        float* __restrict__ out) {          // [512][10]
    const int g = blockIdx.x;
    const int c = threadIdx.x;
    if (c >= N_CLASS) return;
    float s = 0.0f;
    for (int i = 0; i <= N_LAYERS; ++i) {
        s += sb[i * N_CLASS + c];
        const float* pg = pooled + ((size_t)i * N_GRAPHS + g) * HID;
        const float* wi = sw + (size_t)i * HID * N_CLASS;
#pragma unroll 8
        for (int k = 0; k < HID; ++k) s += pg[k] * wi[k * N_CLASS + c];
    }
    out[g * N_CLASS + c] = s;
}

extern "C" void kernel_launch(void* const* d_in, const int* in_sizes, int n_in,
                              void* d_out, int out_size, void* d_ws, size_t ws_size,
                              hipStream_t stream) {
    const float* x       = (const float*)d_in[0];
    const float* n_norm  = (const float*)d_in[1];
    const float* fc1_w   = (const float*)d_in[2];
    const float* fc1_b   = (const float*)d_in[3];
    const float* W1      = (const float*)d_in[4];
    const float* b1      = (const float*)d_in[5];
    const float* W2      = (const float*)d_in[6];
    const float* b2      = (const float*)d_in[7];
    const float* score_w = (const float*)d_in[8];
    const float* score_b = (const float*)d_in[9];
    const int*   esrc    = (const int*)d_in[10];
    const int*   edst    = (const int*)d_in[11];
    const int*   indic   = (const int*)d_in[12];
    float* out = (float*)d_out;

    // workspace layout (floats): hA | hB | agg | pooled[5][512][64]  (~48.6 MB)
    const size_t NH = (size_t)N_NODES * HID;
    float* ws     = (float*)d_ws;
    float* hA     = ws;
    float* hB     = ws + NH;
    float* agg    = ws + 2 * NH;
    float* pooled = ws + 3 * NH;

    k_zero<<<256, 256, 0, stream>>>(pooled, (N_LAYERS + 1) * N_GRAPHS * HID);

    // input projection -> hA, then pool feats[0]
    k_fc1<<<N_NODES / 128, 256, 0, stream>>>(x, fc1_w, fc1_b, hA);
    k_pool<<<(N_NODES * 16) / 256, 256, 0, stream>>>(hA, indic, pooled);

    for (int l = 0; l < N_LAYERS; ++l) {
        // agg = h (self term), then scatter-add neighbor features
        k_copy4<<<2048, 256, 0, stream>>>((float4*)agg, (const float4*)hA, (int)(NH / 4));
        k_edges<<<(N_EDGES * 16) / 256, 256, 0, stream>>>(esrc, edst, hA, agg);
        // fused MLP + graph_norm + relu + residual
        k_mlp<<<N_NODES / 128, 256, 0, stream>>>(agg, hA,
                                                 W1 + (size_t)l * HID * HID, b1 + (size_t)l * HID,
                                                 W2 + (size_t)l * HID * HID, b2 + (size_t)l * HID,
                                                 n_norm, hB);
        k_pool<<<(N_NODES * 16) / 256, 256, 0, stream>>>(hB, indic,
                                                         pooled + (size_t)(l + 1) * N_GRAPHS * HID);
        float* tmp = hA; hA = hB; hB = tmp;
    }

    k_score<<<N_GRAPHS, 32, 0, stream>>>(pooled, score_w, score_b, out);
}